// GraphAffineCoupling_14714557956505
// MI455X (gfx1250) — compile-verified
//
#include <hip/hip_runtime.h>
#include <math.h>

// Problem constants (from reference)
#define B_    2048
#define N_    38
#define E_    4
#define DIN_  9
#define H_    256      // H0 == H1
#define L0_   512
#define EPSV  1e-5f

typedef float v2f __attribute__((ext_vector_type(2)));
typedef float v8f __attribute__((ext_vector_type(8)));

#define WMMA_F32(a, bf, c) \
    __builtin_amdgcn_wmma_f32_16x16x4_f32(false, (a), false, (bf), (short)0, (c), false, false)

// Pipelined K=256 GEMM accumulate: A row pointer (clamped, unconditional loads),
// B row pointer; kh = lane>>4 selects the K pair.
__device__ __forceinline__ v8f gemm_k256(const float* __restrict__ arow,
                                         const float* __restrict__ brow,
                                         int kh, v8f acc)
{
    v2f a0 = *(const v2f*)(arow + 2 * kh);
    v2f b0 = *(const v2f*)(brow + 2 * kh);
#pragma unroll 4
    for (int k0 = 0; k0 < H_ - 4; k0 += 4) {
        v2f a1 = *(const v2f*)(arow + k0 + 4 + 2 * kh);
        v2f b1 = *(const v2f*)(brow + k0 + 4 + 2 * kh);
        acc = WMMA_F32(a0, b0, acc);
        a0 = a1; b0 = b1;
    }
    acc = WMMA_F32(a0, b0, acc);
    return acc;
}

// ---------------------------------------------------------------------------
// Layer 0 graph conv (K = 9, VALU): hA[b,m,c] = hs + hr
// ---------------------------------------------------------------------------
__global__ __launch_bounds__(256)
void k_gconv0(const float* __restrict__ adj, const float* __restrict__ x,
              const float* __restrict__ Ws, const float* __restrict__ bs,
              const float* __restrict__ We, const float* __restrict__ be,
              float* __restrict__ hA)
{
    __shared__ float s_adj[E_ * N_ * N_];   // 5776 floats
    __shared__ float s_m[N_ * 256];         // 9728 floats (64 c-chunk x 4 e)
    __shared__ float s_mx[N_ * DIN_];       // 342 floats
    const int b   = blockIdx.x;
    const int tid = threadIdx.x;

    for (int i = tid; i < N_ * DIN_; i += 256) {
        int n = i / DIN_; int d = i - n * DIN_;
        s_mx[i] = (n & 1) ? x[(b * N_ + n) * DIN_ + d] : 0.0f;   // MASK zeroes even rows
    }
    for (int i = tid; i < E_ * N_ * N_; i += 256)
        s_adj[i] = adj[b * E_ * N_ * N_ + i];
    __syncthreads();

    for (int chunk = 0; chunk < 4; ++chunk) {
        const int c0 = chunk * 64;
        for (int i = tid; i < N_ * 256; i += 256) {
            int n = i >> 8; int jj = i & 255; int j = c0 * 4 + jj;
            float acc = be[j];
            const float* w  = &We[j * DIN_];
            const float* mx = &s_mx[n * DIN_];
#pragma unroll
            for (int d = 0; d < DIN_; ++d) acc += mx[d] * w[d];
            s_m[i] = acc;
        }
        __syncthreads();
        for (int i = tid; i < N_ * 64; i += 256) {
            int m = i >> 6; int cc = i & 63; int c = c0 + cc;
            float hs = bs[c];
            const float* w  = &Ws[c * DIN_];
            const float* mx = &s_mx[m * DIN_];
#pragma unroll
            for (int d = 0; d < DIN_; ++d) hs += mx[d] * w[d];
            float hr = 0.0f;
#pragma unroll
            for (int e = 0; e < E_; ++e) {
                const float* arow = &s_adj[e * N_ * N_ + m * N_];
                const float* mcol = &s_m[cc * 4 + e];
                for (int n = 0; n < N_; ++n) hr += arow[n] * mcol[n * 256];
            }
            hA[(b * N_ + m) * H_ + c] = hs + hr;
        }
        __syncthreads();
    }
}

// ---------------------------------------------------------------------------
// BN stats over width-256 activations (deterministic per-block partials)
// ---------------------------------------------------------------------------
__global__ __launch_bounds__(256)
void k_bnstats(const float* __restrict__ h, float* __restrict__ part)
{
    const int n = blockIdx.x, nb = blockIdx.y, tid = threadIdx.x;
    const int b0 = nb * (B_ / 8);
    float s = 0.f, s2 = 0.f;
    for (int bb = 0; bb < B_ / 8; ++bb) {
        float v = h[((size_t)(b0 + bb) * N_ + n) * H_ + tid];
        s += v; s2 += v * v;
    }
    __shared__ float r0[256], r1[256];
    r0[tid] = s; r1[tid] = s2;
    __syncthreads();
    for (int w = 128; w > 0; w >>= 1) {
        if (tid < w) { r0[tid] += r0[tid + w]; r1[tid] += r1[tid + w]; }
        __syncthreads();
    }
    if (tid == 0) { part[(n * 8 + nb) * 2 + 0] = r0[0]; part[(n * 8 + nb) * 2 + 1] = r1[0]; }
}

__global__ void k_bnfin(const float* __restrict__ part, int nparts, float count,
                        const float* __restrict__ g, const float* __restrict__ bb,
                        float* __restrict__ coef)
{
    int n = threadIdx.x;
    if (n >= N_) return;
    float s = 0.f, s2 = 0.f;
    for (int i = 0; i < nparts; ++i) {
        s  += part[(n * nparts + i) * 2 + 0];
        s2 += part[(n * nparts + i) * 2 + 1];
    }
    float mu  = s / count;
    float var = s2 / count - mu * mu;
    float a   = g[n] * rsqrtf(var + EPSV);
    coef[n * 2] = a; coef[n * 2 + 1] = bb[n] - mu * a;
}

__global__ void k_bnfin3(const float* __restrict__ part, float count,
                         const float* __restrict__ g, const float* __restrict__ bb,
                         float* __restrict__ coef)
{
    int n = threadIdx.x;
    if (n >= N_) return;
    float s = 0.f, s2 = 0.f;
    for (int i = 0; i < B_; ++i) {
        s  += part[((size_t)i * N_ + n) * 2 + 0];
        s2 += part[((size_t)i * N_ + n) * 2 + 1];
    }
    float mu  = s / count;
    float var = s2 / count - mu * mu;
    float a   = g[n] * rsqrtf(var + EPSV);
    coef[n * 2] = a; coef[n * 2 + 1] = bb[n] - mu * a;
}

__global__ __launch_bounds__(256)
void k_bnapply(float* __restrict__ h, const float* __restrict__ coef, int total)
{
    for (int idx = blockIdx.x * 256 + threadIdx.x; idx < total; idx += gridDim.x * 256) {
        int n = (idx >> 8) % N_;    // width is 256
        float v = h[idx] * coef[n * 2] + coef[n * 2 + 1];
        h[idx] = v > 0.f ? v : 0.f;
    }
}

// ---------------------------------------------------------------------------
// Layer 1 graph conv, WMMA f32 16x16x4.  grid (B, 4), 8 waves.
// Phase 2: m-chunk GEMM (38x256)@(256x256) -> LDS (bias be included)
// Phase 3: hs GEMM (K=256) + hr GEMM (K=152 over (e,n)) fused per tile
// Row guards replaced by index clamping: garbage rows are computed but never
// stored, keeping all loads unconditional (no EXEC save/restore in hot loop).
// ---------------------------------------------------------------------------
__global__ __launch_bounds__(256)
void k_gconv1(const float* __restrict__ adj, const float* __restrict__ h1,
              const float* __restrict__ Ws, const float* __restrict__ bs,
              const float* __restrict__ We, const float* __restrict__ be,
              float* __restrict__ hB)
{
    __shared__ float s_adj[E_ * N_ * N_];   // 23104 B
    __shared__ float s_m[N_ * 256];         // 38912 B
    const int b     = blockIdx.x;
    const int chunk = blockIdx.y;           // 0..3
    const int tid   = threadIdx.x;
    const int wave  = tid >> 5;
    const int lane  = tid & 31;
    const int l15   = lane & 15;
    const int kh    = lane >> 4;

    for (int i = tid; i < E_ * N_ * N_; i += 256)
        s_adj[i] = adj[b * E_ * N_ * N_ + i];
    __syncthreads();

    const int jbase = chunk * 256;

    // ---- Phase 2: 48 tiles (3 mt x 16 nt), 6 per wave ----
    for (int t6 = 0; t6 < 6; ++t6) {
        int t  = wave * 6 + t6;
        int mt = t >> 4, nt = t & 15;
        int m  = mt * 16 + l15;
        int mc = m < N_ ? m : N_ - 1;       // clamp: rows >= 38 are never stored
        const float* arow = &h1[((size_t)b * N_ + mc) * H_];
        int wrow = jbase + nt * 16 + l15;
        const float* brow = &We[(size_t)wrow * H_];
        v8f acc = {0.f, 0.f, 0.f, 0.f, 0.f, 0.f, 0.f, 0.f};
        acc = gemm_k256(arow, brow, kh, acc);
        float bias = be[wrow];
        int col = nt * 16 + l15;
#pragma unroll
        for (int j = 0; j < 8; ++j) {
            int row = mt * 16 + j + kh * 8;
            if (row < N_) s_m[row * 256 + col] = acc[j] + bias;
        }
    }
    __syncthreads();

    // ---- Phase 3: 12 tiles (3 mt x 4 ct); waves 0..7 then 0..3 again ----
    for (int rep = 0; rep < 2; ++rep) {
        int id = (rep == 0) ? wave : ((wave < 4) ? 8 + wave : -1);
        if (id < 0) continue;               // uniform per wave
        int mt = id >> 2, ct = id & 3;
        int m  = mt * 16 + l15;
        int mc = m < N_ ? m : N_ - 1;
        const float* arow = &h1[((size_t)b * N_ + mc) * H_];
        int c = chunk * 64 + ct * 16 + l15;
        const float* brow = &Ws[(size_t)c * H_];
        v8f acc = {0.f, 0.f, 0.f, 0.f, 0.f, 0.f, 0.f, 0.f};
        acc = gemm_k256(arow, brow, kh, acc);           // hs part
        // hr: K = 152 over k = e*38 + n  (LDS operands)
        int cc = ct * 16 + l15;
        for (int k0 = 0; k0 < E_ * N_; k0 += 4) {
            int k  = k0 + 2 * kh;
            int e0 = k / N_,  n0 = k - N_ * e0;
            int k1 = k + 1;
            int e1 = k1 / N_, n1 = k1 - N_ * e1;
            v2f a;
            a.x = s_adj[e0 * (N_ * N_) + mc * N_ + n0];
            a.y = s_adj[e1 * (N_ * N_) + mc * N_ + n1];
            v2f bf;
            bf.x = s_m[n0 * 256 + cc * 4 + e0];
            bf.y = s_m[n1 * 256 + cc * 4 + e1];
            acc = WMMA_F32(a, bf, acc);
        }
        float bias = bs[c];
#pragma unroll
        for (int j = 0; j < 8; ++j) {
            int row = mt * 16 + j + kh * 8;
            if (row < N_) hB[((size_t)b * N_ + row) * H_ + c] = acc[j] + bias;
        }
    }
}

// ---------------------------------------------------------------------------
// Linear (512) statistics pass, WMMA: h3 = h @ lW0^T + lb0 per batch,
// staged per 256-column half in LDS; deterministic per-n sum/sumsq.
// ---------------------------------------------------------------------------
__global__ __launch_bounds__(256)
void k_linstats(const float* __restrict__ h, const float* __restrict__ W,
                const float* __restrict__ bias, float* __restrict__ part)
{
    __shared__ float s_h4[N_ * 256];
    __shared__ float s_p[256 * 2];
    const int b = blockIdx.x, tid = threadIdx.x;
    const int wave = tid >> 5, lane = tid & 31, l15 = lane & 15, kh = lane >> 4;
    float s = 0.f, s2 = 0.f;

    for (int half = 0; half < 2; ++half) {
        const int jb = half * 256;
        for (int t6 = 0; t6 < 6; ++t6) {            // 48 tiles, 6 per wave
            int t  = wave * 6 + t6;
            int mt = t >> 4, nt = t & 15;
            int m  = mt * 16 + l15;
            int mc = m < N_ ? m : N_ - 1;
            const float* arow = &h[((size_t)b * N_ + mc) * H_];
            int jj = jb + nt * 16 + l15;
            const float* brow = &W[(size_t)jj * H_];
            v8f acc = {0.f, 0.f, 0.f, 0.f, 0.f, 0.f, 0.f, 0.f};
            acc = gemm_k256(arow, brow, kh, acc);
            float bb = bias[jj];
            int col = nt * 16 + l15;
#pragma unroll
            for (int j = 0; j < 8; ++j) {
                int row = mt * 16 + j + kh * 8;
                if (row < N_) s_h4[row * 256 + col] = acc[j] + bb;
            }
        }
        __syncthreads();
        if (tid < 228) {                            // 6 threads per node, fixed order
            int n = tid % N_, g = tid / N_;
            const float* hr = &s_h4[n * 256];
            for (int j = g; j < 256; j += 6) { float v = hr[j]; s += v; s2 += v * v; }
        }
        __syncthreads();                            // before next half overwrites s_h4
    }
    s_p[tid * 2] = s; s_p[tid * 2 + 1] = s2;
    __syncthreads();
    if (tid < N_) {
        float ss = 0.f, ss2 = 0.f;
        for (int g = 0; g < 6; ++g) {
            ss  += s_p[(tid + N_ * g) * 2 + 0];
            ss2 += s_p[(tid + N_ * g) * 2 + 1];
        }
        part[((size_t)b * N_ + tid) * 2 + 0] = ss;
        part[((size_t)b * N_ + tid) * 2 + 1] = ss2;
    }
}

// ---------------------------------------------------------------------------
// Final: WMMA h3 recompute, BN3+ReLU, head GEMM to 18, coupling + logdet
// ---------------------------------------------------------------------------
__global__ __launch_bounds__(256)
void k_final(const float* __restrict__ h, const float* __restrict__ W,
             const float* __restrict__ lb, const float* __restrict__ coef3,
             const float* __restrict__ fW, const float* __restrict__ fb,
             const float* __restrict__ x, float* __restrict__ out,
             float* __restrict__ logdet)
{
    __shared__ float s_h4[N_ * 256];
    __shared__ float s_f[N_ * 18];
    __shared__ float s_ls[N_ * DIN_];
    const int b = blockIdx.x, tid = threadIdx.x;
    const int wave = tid >> 5, lane = tid & 31, l15 = lane & 15, kh = lane >> 4;
    for (int i = tid; i < N_ * 18; i += 256) s_f[i] = 0.f;
    __syncthreads();

    for (int half = 0; half < 2; ++half) {
        const int jb = half * 256;
        for (int t6 = 0; t6 < 6; ++t6) {
            int t  = wave * 6 + t6;
            int mt = t >> 4, nt = t & 15;
            int m  = mt * 16 + l15;
            int mc = m < N_ ? m : N_ - 1;
            const float* arow = &h[((size_t)b * N_ + mc) * H_];
            int jj = jb + nt * 16 + l15;
            const float* brow = &W[(size_t)jj * H_];
            v8f acc = {0.f, 0.f, 0.f, 0.f, 0.f, 0.f, 0.f, 0.f};
            acc = gemm_k256(arow, brow, kh, acc);
            float bb = lb[jj];
            int col = nt * 16 + l15;
#pragma unroll
            for (int j = 0; j < 8; ++j) {
                int row = mt * 16 + j + kh * 8;
                if (row < N_) {
                    float v = (acc[j] + bb) * coef3[row * 2] + coef3[row * 2 + 1];
                    s_h4[row * 256 + col] = v > 0.f ? v : 0.f;
                }
            }
        }
        __syncthreads();
        for (int o = tid; o < N_ * 18; o += 256) {
            int n = o / 18, q = o - n * 18;
            const float* h4 = &s_h4[n * 256];
            const float* w  = &fW[(size_t)q * L0_ + jb];
            float acc = 0.f;
            for (int k = 0; k < 256; ++k) acc += h4[k] * w[k];
            s_f[o] += acc;
        }
        __syncthreads();
    }

    for (int o = tid; o < N_ * DIN_; o += 256) {
        int n = o / DIN_, d = o - n * DIN_;
        float sl = s_f[n * 18 + d]     + fb[d];
        float tv = s_f[n * 18 + 9 + d] + fb[9 + d];
        float xv = x[((size_t)b * N_ + n) * DIN_ + d];
        float sg = 1.f / (1.f + expf(-sl));
        out[((size_t)b * N_ + n) * DIN_ + d] = (n & 1) ? xv : (xv + tv) * sg;
        s_ls[o] = (sl >= 0.f) ? -log1pf(expf(-sl)) : (sl - log1pf(expf(sl)));
    }
    __syncthreads();
    if (tid == 0) {
        float acc = 0.f;
        for (int i = 0; i < N_ * DIN_; ++i) acc += s_ls[i];
        logdet[b] = acc;
    }
}

// ---------------------------------------------------------------------------
extern "C" void kernel_launch(void* const* d_in, const int* in_sizes, int n_in,
                              void* d_out, int out_size, void* d_ws, size_t ws_size,
                              hipStream_t stream)
{
    (void)in_sizes; (void)n_in; (void)out_size; (void)ws_size;
    const float* adj  = (const float*)d_in[0];
    const float* x    = (const float*)d_in[1];
    const float* cWs0 = (const float*)d_in[2];
    const float* cbs0 = (const float*)d_in[3];
    const float* cWe0 = (const float*)d_in[4];
    const float* cbe0 = (const float*)d_in[5];
    const float* cg0  = (const float*)d_in[6];
    const float* cb0  = (const float*)d_in[7];
    const float* cWs1 = (const float*)d_in[8];
    const float* cbs1 = (const float*)d_in[9];
    const float* cWe1 = (const float*)d_in[10];
    const float* cbe1 = (const float*)d_in[11];
    const float* cg1  = (const float*)d_in[12];
    const float* cb1  = (const float*)d_in[13];
    const float* lW0  = (const float*)d_in[14];
    const float* lb0  = (const float*)d_in[15];
    const float* lg0  = (const float*)d_in[16];
    const float* lbb0 = (const float*)d_in[17];
    const float* fW   = (const float*)d_in[18];
    const float* fb   = (const float*)d_in[19];

    float* out    = (float*)d_out;
    float* logdet = out + (size_t)B_ * N_ * DIN_;

    char* ws = (char*)d_ws;
    float* hA    = (float*)ws; ws += (size_t)B_ * N_ * H_ * 4;
    float* hB    = (float*)ws; ws += (size_t)B_ * N_ * H_ * 4;
    float* part0 = (float*)ws; ws += N_ * 8 * 2 * 4;
    float* part1 = (float*)ws; ws += N_ * 8 * 2 * 4;
    float* coef0 = (float*)ws; ws += N_ * 2 * 4;
    float* coef1 = (float*)ws; ws += N_ * 2 * 4;
    float* coef3 = (float*)ws; ws += N_ * 2 * 4;
    float* part3 = (float*)ws; ws += (size_t)B_ * N_ * 2 * 4;

    k_gconv0<<<B_, 256, 0, stream>>>(adj, x, cWs0, cbs0, cWe0, cbe0, hA);
    k_bnstats<<<dim3(N_, 8), 256, 0, stream>>>(hA, part0);
    k_bnfin<<<1, 64, 0, stream>>>(part0, 8, (float)(B_ * H_), cg0, cb0, coef0);
    k_bnapply<<<4096, 256, 0, stream>>>(hA, coef0, B_ * N_ * H_);
    k_gconv1<<<dim3(B_, 4), 256, 0, stream>>>(adj, hA, cWs1, cbs1, cWe1, cbe1, hB);
    k_bnstats<<<dim3(N_, 8), 256, 0, stream>>>(hB, part1);
    k_bnfin<<<1, 64, 0, stream>>>(part1, 8, (float)(B_ * H_), cg1, cb1, coef1);
    k_bnapply<<<4096, 256, 0, stream>>>(hB, coef1, B_ * N_ * H_);
    k_linstats<<<B_, 256, 0, stream>>>(hB, lW0, lb0, part3);
    k_bnfin3<<<1, 64, 0, stream>>>(part3, (float)(B_ * L0_), lg0, lbb0, coef3);
    k_final<<<B_, 256, 0, stream>>>(hB, lW0, lb0, coef3, fW, fb, x, out, logdet);
}